// Elman_64141041599040
// MI455X (gfx1250) — compile-verified
//
#include <hip/hip_runtime.h>

typedef float v2f __attribute__((ext_vector_type(2)));
typedef float v4f __attribute__((ext_vector_type(4)));
typedef float v8f __attribute__((ext_vector_type(8)));
typedef int   v4i __attribute__((vector_size(16)));

#define AS1 __attribute__((address_space(1)))
#define AS3 __attribute__((address_space(3)))

#define KDIM 300          // inner dim (INSIZE == HSIZE == OUTSIZE == 300)
#define KP   304          // padded to 19*16
#define KP2  (2 * KP)     // 608: row stride of k-pair-interleaved BT layout
#define NDIM 300
#define BATCH 128
#define TSTEPS 1024
#define ROWS_TOTAL (BATCH * TSTEPS)   // 131072

#if defined(__has_builtin)
#if __has_builtin(__builtin_amdgcn_global_load_async_to_lds_b128)
#define HAVE_ASYNC_LDS 1
#endif
#endif

__device__ __forceinline__ v8f wmma_f32(v2f a, v2f b, v8f c) {
    // V_WMMA_F32_16X16X4_F32 : D = A(16x4) * B(4x16) + C(16x16)
    return __builtin_amdgcn_wmma_f32_16x16x4_f32(
        /*neg_a=*/false, a, /*neg_b=*/false, b,
        /*c_mod=*/(short)0, c, /*reuse_a=*/false, /*reuse_b=*/false);
}

__device__ __forceinline__ void wait_async_zero() {
#if defined(HAVE_ASYNC_LDS)
#if __has_builtin(__builtin_amdgcn_s_wait_asynccnt)
    __builtin_amdgcn_s_wait_asynccnt(0);
#else
    asm volatile("s_wait_asynccnt 0x0" ::: "memory");
#endif
#endif
}

// ---------------------------------------------------------------------------
// Kernel 0: transpose + zero-pad weights into ws, in k-pair interleaved
// layout:  BTp[(k>>1)*KP2 + 2*n + (k&1)] = W^T[k][n]   (zero outside 300x300)
// so a lane fetches {B[k][n], B[k+1][n]} with ONE aligned global_load_b64.
//   ws[0*KP*KP] = W1x^T   (W1[:, :300])
//   ws[1*KP*KP] = W1h^T   (W1[:, 300:])
//   ws[2*KP*KP] = W2^T
// ---------------------------------------------------------------------------
__global__ void prep_weights(const float* __restrict__ W1,
                             const float* __restrict__ W2,
                             float* __restrict__ ws) {
    int o = blockIdx.x * blockDim.x + threadIdx.x;
    int m = blockIdx.y;
    if (o >= KP * KP) return;
    int kp  = o / KP2;
    int rem = o - kp * KP2;
    int n = rem >> 1;
    int k = 2 * kp + (rem & 1);
    float v = 0.0f;
    if (k < KDIM && n < NDIM) {
        if (m == 0)      v = W1[n * (2 * KDIM) + k];
        else if (m == 1) v = W1[n * (2 * KDIM) + KDIM + k];
        else             v = W2[n * KDIM + k];
    }
    ws[m * KP * KP + o] = v;
}

// ---------------------------------------------------------------------------
// Generic D[r,n] = bias[n] + sum_k A[r,k] * BT[k][n], BT in pair layout.
// 128 threads (4 waves), 32 rows per workgroup staged in LDS via async-to-LDS.
// Safe for A == D (all reads of A happen before the barrier; stores after).
// Each wave: row-tile = wave>>1, column-tile half = wave&1.
// ---------------------------------------------------------------------------
__global__ __launch_bounds__(128) void gemm300(
        const float* A, const float* __restrict__ BT,
        const float* __restrict__ bias, float* D) {
    __shared__ alignas(16) float As[32 * KP];
    const int tid = threadIdx.x;
    const int rowStart = blockIdx.x * 32;

    // Stage 32 rows of A (zero-padded K) into LDS, 16B chunks per lane.
    // 32 rows * 76 chunks = 2432 -> exactly 19 iterations for 128 threads.
    for (int idx = tid; idx < 32 * (KP / 4); idx += 128) {
        int r  = idx / (KP / 4);
        int cc = idx - r * (KP / 4);
        float* dst = &As[r * KP + cc * 4];
        if (cc < KDIM / 4) {
            const float* src = &A[(rowStart + r) * KDIM + cc * 4];
#if defined(HAVE_ASYNC_LDS)
            __builtin_amdgcn_global_load_async_to_lds_b128(
                (AS1 v4i*)src, (AS3 v4i*)dst, 0, 0);
#else
            *(v4f*)dst = *(const v4f*)src;
#endif
        } else {
            v4f z = {0.0f, 0.0f, 0.0f, 0.0f};
            *(v4f*)dst = z;
        }
    }
    wait_async_zero();
    __syncthreads();

    const int wave = tid >> 5;
    const int lane = tid & 31;
    const int rt   = wave >> 1;               // row tile 0..1
    const int nt0  = (wave & 1) ? 10 : 0;     // column tile range
    const int nt1  = (wave & 1) ? 19 : 10;
    const int hl   = lane >> 4;               // half-wave: K offset +2
    const int nlo  = lane & 15;

    const float* Arow = &As[(rt * 16 + nlo) * KP + hl * 2];

    for (int nt = nt0; nt < nt1; ++nt) {
        const int n = nt * 16 + nlo;
        const bool nvalid = (n < NDIM);
        float bv = nvalid ? bias[n] : 0.0f;
        v8f c = {bv, bv, bv, bv, bv, bv, bv, bv};
        // pair layout: {B[k0+2hl][n], B[k0+2hl+1][n]} in one b64
        const float* Bp = &BT[hl * KP2 + 2 * n];
        for (int k0 = 0; k0 < KP; k0 += 4) {
            v2f a = *(const v2f*)(Arow + k0);                 // ds_load_b64
            v2f b = *(const v2f*)(Bp + (k0 >> 1) * KP2);      // global b64
            c = wmma_f32(a, b, c);
        }
        if (nvalid) {
#pragma unroll
            for (int j = 0; j < 8; ++j) {
                int m = j + hl * 8;                           // C/D row layout
                D[(rowStart + rt * 16 + m) * NDIM + n] = c[j];
            }
        }
    }
}

// ---------------------------------------------------------------------------
// Recurrence: h_t = sigmoid(P[:,t,:] + h_{t-1} @ W1h^T), P preloaded in dout.
// 8 blocks x 16 batch rows; 19 waves, each owns one 16-wide column tile.
// h double-buffered in LDS (zero-padded cols 300..303), one barrier per step.
// Overwrites dout(b,t,:) with h_t; writes h_final past the outs region.
// ---------------------------------------------------------------------------
__global__ __launch_bounds__(608) void elman_recur(
        const float* __restrict__ W1hT, float* dout) {
    __shared__ alignas(16) float hb[2][16 * KP];
    const int tid  = threadIdx.x;
    const int wave = tid >> 5;
    const int lane = tid & 31;
    const int r0   = blockIdx.x * 16;        // batch rows r0 .. r0+15

    for (int i = tid; i < 16 * KP; i += 608) hb[0][i] = 0.0f;  // h0 = 0
    __syncthreads();

    const int hl  = lane >> 4;
    const int nlo = lane & 15;
    const int n   = wave * 16 + nlo;         // 0..303
    const bool nvalid = (n < NDIM);
    const float* Brow = &W1hT[hl * KP2 + 2 * n];   // pair layout

    for (int t = 0; t < TSTEPS; ++t) {
        const int cur = t & 1;
        const float* hcur = hb[cur];
        float* hnext      = hb[cur ^ 1];

        // C = pre-activation tile from dout (x@W1x^T + b1)
        v8f c;
#pragma unroll
        for (int j = 0; j < 8; ++j) {
            int m = j + hl * 8;
            c[j] = nvalid ? dout[((r0 + m) * TSTEPS + t) * NDIM + n] : 0.0f;
        }

        // += h_{t-1} @ W1h^T
        const float* Arow = &hcur[nlo * KP + hl * 2];
        for (int k0 = 0; k0 < KP; k0 += 4) {
            v2f a = *(const v2f*)(Arow + k0);                 // ds_load_b64
            v2f b = *(const v2f*)(Brow + (k0 >> 1) * KP2);    // global b64
            c = wmma_f32(a, b, c);
        }

        // sigmoid, store h_t to LDS (next step's A) and to dout (in place)
#pragma unroll
        for (int j = 0; j < 8; ++j) {
            int m = j + hl * 8;
            float s = 1.0f / (1.0f + __expf(-c[j]));
            hnext[m * KP + n] = nvalid ? s : 0.0f;            // keep K padding 0
            if (nvalid) {
                dout[((r0 + m) * TSTEPS + t) * NDIM + n] = s;
                if (t == TSTEPS - 1)
                    dout[ROWS_TOTAL * NDIM + (r0 + m) * NDIM + n] = s;
            }
        }
        __syncthreads();
    }
}

// ---------------------------------------------------------------------------
extern "C" void kernel_launch(void* const* d_in, const int* in_sizes, int n_in,
                              void* d_out, int out_size, void* d_ws, size_t ws_size,
                              hipStream_t stream) {
    const float* x  = (const float*)d_in[0];   // (128,1024,300)
    const float* W1 = (const float*)d_in[1];   // (300,600)
    const float* b1 = (const float*)d_in[2];   // (300,)
    const float* W2 = (const float*)d_in[3];   // (300,300)
    const float* b2 = (const float*)d_in[4];   // (300,)
    float* out = (float*)d_out;                // outs (B,T,300) ++ h_final (B,300)
    float* wsf = (float*)d_ws;                 // 3 * KP*KP floats = ~1.06 MB

    float* W1xT = wsf;
    float* W1hT = wsf + KP * KP;
    float* W2T  = wsf + 2 * KP * KP;

    // Phase 0: transpose + pad weights into pair layout (tiny, L2 resident)
    dim3 pg((KP * KP + 255) / 256, 3);
    prep_weights<<<pg, 256, 0, stream>>>(W1, W2, wsf);

    // Phase 1: P = x @ W1x^T + b1  -> d_out (fully parallel GEMM)
    gemm300<<<ROWS_TOTAL / 32, 128, 0, stream>>>(x, W1xT, b1, out);

    // Phase 2: sequential recurrence over T, 8 independent batch-row chains
    elman_recur<<<BATCH / 16, 608, 0, stream>>>(W1hT, out);

    // Phase 3: out = h @ W2^T + b2, in place on d_out (fully parallel GEMM)
    gemm300<<<ROWS_TOTAL / 32, 128, 0, stream>>>(out, W2T, b2, out);
}